// TaskSpecificLinear_24910810316924
// MI455X (gfx1250) — compile-verified
//
#include <hip/hip_runtime.h>

typedef float v2f __attribute__((ext_vector_type(2)));
typedef float v8f __attribute__((ext_vector_type(8)));

constexpr int kN   = 2048;   // samples
constexpr int kIn  = 256;    // in_size (K)
constexpr int kOut = 256;    // out_size
constexpr int kT   = 128;    // tasks
constexpr int APAD = 258;    // LDS row stride (even, avoids 64-bank conflicts)

// d_ws int32 layout:
//  [0   .. 128)   counts[t]
//  [128 .. 257)   offsets[t] (exclusive scan, T+1 entries)
//  [257 .. 385)   cursor[t]  (scatter write position)
//  [512 .. 2560)  perm[n]    (sample ids sorted by task)
#define WS_COUNTS  0
#define WS_OFFSETS 128
#define WS_CURSOR  257
#define WS_PERM    512

__global__ void tsl_zero_counts(int* ws) {
    int i = threadIdx.x;
    if (i < kT) ws[WS_COUNTS + i] = 0;
}

__global__ void tsl_count(const int* __restrict__ task_ids, int* ws) {
    int n = blockIdx.x * blockDim.x + threadIdx.x;
    if (n < kN) atomicAdd(&ws[WS_COUNTS + task_ids[n]], 1);
}

__global__ void tsl_scan(int* ws) {
    if (threadIdx.x == 0) {
        int acc = 0;
        for (int t = 0; t < kT; ++t) {
            ws[WS_OFFSETS + t] = acc;
            ws[WS_CURSOR  + t] = acc;
            acc += ws[WS_COUNTS + t];
        }
        ws[WS_OFFSETS + kT] = acc;
    }
}

__global__ void tsl_scatter(const int* __restrict__ task_ids, int* ws) {
    int n = blockIdx.x * blockDim.x + threadIdx.x;
    if (n < kN) {
        int t = task_ids[n];
        int pos = atomicAdd(&ws[WS_CURSOR + t], 1);
        ws[WS_PERM + pos] = n;
    }
}

// One block per task. 256 threads = 8 wave32; wave w owns output cols
// [w*32, w*32+32) as two 16-wide N tiles. Samples of the task are processed
// in chunks of 16 rows (one WMMA M tile), A staged in LDS.
__global__ __launch_bounds__(256)
void tsl_gemm(const float* __restrict__ x, const float* __restrict__ W,
              const int* __restrict__ ws, float* __restrict__ out) {
    __shared__ float As[16 * APAD];
    __shared__ int   s_samp[16];

    const int t   = blockIdx.x;
    const int beg = ws[WS_OFFSETS + t];
    const int end = ws[WS_OFFSETS + t + 1];
    if (beg == end) return;

    const int* perm = ws + WS_PERM;
    const int tid   = threadIdx.x;
    const int lane  = tid & 31;
    const int wave  = tid >> 5;
    const int lidx  = lane & 15;     // N (or M) index within tile
    const int lhalf = lane >> 4;     // selects K pair / M+8 half

    const float* Wt = W + (size_t)t * kIn * kOut;
    const int col0 = wave * 32 + lidx;   // N tile 0 column
    const int col1 = col0 + 16;          // N tile 1 column

    for (int base = beg; base < end; base += 16) {
        const int valid = (end - base) < 16 ? (end - base) : 16;

        // ---- stage chunk sample ids + A tile (16 x 256) into LDS ----
        if (tid < 16) s_samp[tid] = (tid < valid) ? perm[base + tid] : 0;
        {
            const int row  = tid >> 4;          // 0..15
            const int kblk = (tid & 15) << 4;   // 0,16,...,240
            float4 v0, v1, v2, v3;
            if (row < valid) {
                const float* src = x + (size_t)perm[base + row] * kIn + kblk;
                const float4* s4 = (const float4*)src;
                v0 = s4[0]; v1 = s4[1]; v2 = s4[2]; v3 = s4[3];
            } else {
                v0 = v1 = v2 = v3 = make_float4(0.f, 0.f, 0.f, 0.f);
            }
            float* dst = &As[row * APAD + kblk];
            dst[0]  = v0.x; dst[1]  = v0.y; dst[2]  = v0.z; dst[3]  = v0.w;
            dst[4]  = v1.x; dst[5]  = v1.y; dst[6]  = v1.z; dst[7]  = v1.w;
            dst[8]  = v2.x; dst[9]  = v2.y; dst[10] = v2.z; dst[11] = v2.w;
            dst[12] = v3.x; dst[13] = v3.y; dst[14] = v3.z; dst[15] = v3.w;
        }
        __syncthreads();

        // ---- K loop: 64 steps of V_WMMA_F32_16X16X4_F32, 2 N tiles ----
        v8f acc0 = {}; v8f acc1 = {};
        const float* As_row = &As[lidx * APAD];
        #pragma unroll 4
        for (int kb = 0; kb < kIn; kb += 4) {
            const int k0 = kb + 2 * lhalf;      // this lane's K pair
            v2f a;                              // A: M=lidx, K=k0,k0+1
            a.x = As_row[k0];
            a.y = As_row[k0 + 1];
            const float* wp = Wt + (size_t)k0 * kOut;
            v2f b0, b1;                         // B: K=k0,k0+1 ; N=col
            b0.x = wp[col0];        b1.x = wp[col1];
            b0.y = wp[kOut + col0]; b1.y = wp[kOut + col1];
            acc0 = __builtin_amdgcn_wmma_f32_16x16x4_f32(
                false, a, false, b0, (short)0, acc0, false, false);
            acc1 = __builtin_amdgcn_wmma_f32_16x16x4_f32(
                false, a, false, b1, (short)0, acc1, false, false);
        }

        // ---- store D: lane l, vgpr r -> row M = r + 8*lhalf, col = N ----
        #pragma unroll
        for (int r = 0; r < 8; ++r) {
            const int m = r + 8 * lhalf;
            if (m < valid) {
                float* orow = out + (size_t)s_samp[m] * kOut;
                orow[col0] = acc0[r];
                orow[col1] = acc1[r];
            }
        }
        __syncthreads();
    }
}

extern "C" void kernel_launch(void* const* d_in, const int* in_sizes, int n_in,
                              void* d_out, int out_size, void* d_ws, size_t ws_size,
                              hipStream_t stream) {
    const float* x        = (const float*)d_in[0];
    const int*   task_ids = (const int*)d_in[1];
    const float* W        = (const float*)d_in[2];
    float*       out      = (float*)d_out;
    int*         ws       = (int*)d_ws;

    tsl_zero_counts<<<1, 128, 0, stream>>>(ws);
    tsl_count<<<(kN + 255) / 256, 256, 0, stream>>>(task_ids, ws);
    tsl_scan<<<1, 32, 0, stream>>>(ws);
    tsl_scatter<<<(kN + 255) / 256, 256, 0, stream>>>(task_ids, ws);
    tsl_gemm<<<kT, 256, 0, stream>>>(x, W, ws, out);
}